// GroupEquivariantDepthwiseConv_89404039233992
// MI455X (gfx1250) — compile-verified
//
#include <hip/hip_runtime.h>
#include <cstdint>

// GroupEquivariantDepthwiseConv (C4) for MI455X / gfx1250.
// Memory-bound: 201 MB in + 805 MB out => ~43us floor @ 23.3 TB/s.
// Strategy: async global->LDS B128 staging (ASYNCcnt), compute all 4
// rotations per staged tile, non-temporal B128 stores.

constexpr int CC = 192, HH = 128, WW = 128;
constexpr int TILE_H = 8;
constexpr int LDS_ROWS = TILE_H + 2;

typedef float v4f __attribute__((ext_vector_type(4)));
typedef int   v4i __attribute__((ext_vector_type(4)));

// Rotated-kernel accessor: rot90(w, r) in the (kh,kw) plane, CCW like numpy.
//  r=0: w[i][j]  r=1: w[j][2-i]  r=2: w[2-i][2-j]  r=3: w[2-j][i]
__device__ __forceinline__ float rotw(const float wk[9], int r, int i, int j) {
  int idx;
  if      (r == 0) idx = i * 3 + j;
  else if (r == 1) idx = j * 3 + (2 - i);
  else if (r == 2) idx = (2 - i) * 3 + (2 - j);
  else             idx = (2 - j) * 3 + i;
  return wk[idx];  // fully unrolled call sites -> constant idx -> registers
}

__global__ __launch_bounds__(256) void geq_dw_c4_kernel(
    const float* __restrict__ x,     // [B, C, H, W]
    const float* __restrict__ wgt,   // [C, 1, 3, 3]
    float* __restrict__ out) {       // [B, 4C, H, W]
  __shared__ __align__(16) float tile[LDS_ROWS * WW];

  const int tid  = threadIdx.x;
  const int lane = tid & 31;   // wave32
  const int wave = tid >> 5;   // 8 waves / block

  const int tpi = HH / TILE_H;            // 16 tiles per (b,c) plane
  const int bid = blockIdx.x;
  const int t   = bid % tpi;
  const int bc  = bid / tpi;              // b*C + c
  const int c   = bc % CC;
  const int b   = bc / CC;
  const int h0  = t * TILE_H;

  // ---- stage rows h0-1 .. h0+TILE_H into LDS ---------------------------
  // One image row = 128 floats = 512 B = one wave32 x b128 async load.
  const float* src = x + (size_t)bc * (HH * WW);
  for (int g = wave; g < LDS_ROWS; g += 8) {
    const int hr = h0 - 1 + g;
    float* lrow = tile + g * WW + lane * 4;
    if (hr >= 0 && hr < HH) {
      const float* gp = src + (size_t)hr * WW + lane * 4;
#if __has_builtin(__builtin_amdgcn_global_load_async_to_lds_b128)
      v4i* gpi = (v4i*)gp;     // drop const, retype to 16B vector
      v4i* lpi = (v4i*)lrow;
      __builtin_amdgcn_global_load_async_to_lds_b128(
          (__attribute__((address_space(1))) v4i*)gpi,
          (__attribute__((address_space(3))) v4i*)lpi,
          /*offset=*/0, /*cpol=*/0);
#else
      *(v4f*)lrow = *(const v4f*)gp;
#endif
    } else {
      v4f z = {0.f, 0.f, 0.f, 0.f};   // zero-pad halo rows (h=-1, h=128)
      *(v4f*)lrow = z;
    }
  }

#if __has_builtin(__builtin_amdgcn_s_wait_asynccnt)
  __builtin_amdgcn_s_wait_asynccnt(0);
#elif defined(__gfx1250__)
  asm volatile("s_wait_asynccnt 0" ::: "memory");
#endif
  __syncthreads();

  // ---- per-channel 3x3 weights (block-uniform) -------------------------
  const float* wp = wgt + c * 9;
  float wk[9];
#pragma unroll
  for (int k = 0; k < 9; ++k) wk[k] = wp[k];

  // ---- compute: wave -> output row, lane -> 4 consecutive pixels -------
  const int rr = wave;          // row within tile
  const int wb = lane * 4;      // first output column
  v4f acc[4] = {};              // 4 rotations x 4 pixels

#pragma unroll
  for (int i = 0; i < 3; ++i) {
    const float* row = tile + (rr + i) * WW;          // x row (h + i - 1)
    const float tl = (wb > 0)       ? row[wb - 1] : 0.f;
    const v4f   m  = *(const v4f*)(row + wb);
    const float tr = (wb + 4 < WW)  ? row[wb + 4] : 0.f;
    // sliding windows: s_j[p] = x[col wb+p+j-1]
    const v4f s0 = {tl, m.x, m.y, m.z};
    const v4f s1 = m;
    const v4f s2 = {m.y, m.z, m.w, tr};
#pragma unroll
    for (int r = 0; r < 4; ++r) {
      acc[r] += rotw(wk, r, i, 0) * s0;   // fp-contract=fast -> v_pk_fma_f32
      acc[r] += rotw(wk, r, i, 1) * s1;
      acc[r] += rotw(wk, r, i, 2) * s2;
    }
  }

  // ---- non-temporal coalesced B128 stores ------------------------------
  const int h = h0 + rr;
#pragma unroll
  for (int r = 0; r < 4; ++r) {
    const size_t off =
        ((((size_t)b * 4 + r) * CC + c) * (size_t)HH + h) * WW + wb;
    __builtin_nontemporal_store(acc[r], (v4f*)(out + off));
  }
}

extern "C" void kernel_launch(void* const* d_in, const int* in_sizes, int n_in,
                              void* d_out, int out_size, void* d_ws, size_t ws_size,
                              hipStream_t stream) {
  const float* x   = (const float*)d_in[0];   // [16,192,128,128] fp32
  const float* wgt = (const float*)d_in[1];   // [192,1,3,3] fp32
  float* out       = (float*)d_out;           // [16,768,128,128] fp32

  constexpr int B = 16;
  const int blocks = B * CC * (HH / TILE_H);  // 49152 blocks, 256 thr each
  geq_dw_c4_kernel<<<blocks, 256, 0, stream>>>(x, wgt, out);
}